// CMA_16535624089790
// MI455X (gfx1250) — compile-verified
//
#include <hip/hip_runtime.h>

// CMA adaptive equalizer, block-exact reformulation for CDNA5 (gfx1250).
//
// Math (identical to the reference scan up to fp reassociation):
//   v(n) = <w_blk, u_n> + sum_{m<n} c(m) * G[m,n]
//   c(m) = 2*LR*sign(R2-|v(m)|^2) * v(m),   G[m,n] = sum ubar_m u_n
//   w   += sum_m c(m) * conj(u_m)           (end of each 16-sample block)
//
// The Gram tiles G are input-only (w-independent), so they are computed by a
// GPU-wide parallel kernel (8192 independent blocks, 60 f32-WMMAs each) into
// d_ws; the serial wave then only runs the w-dependent filter GEMM (30 WMMAs)
// + the 16-step recurrence per block, with G tiles hitting in the 192MB L2.
// If d_ws is too small, fall back to the monolithic single-kernel version.

typedef __attribute__((ext_vector_type(2))) float v2f;
typedef __attribute__((ext_vector_type(8))) float v8f;

#define NSAMP 131072
#define QF    30            // TAPS*DIMS
#define BLK   16
#define NBLK  (NSAMP / BLK)
#define LSTR  128           // LDS row stride (floats), 120 used
#define R2CMA 1.32f
#define MU    0x1p-13f      // 2 * LR = 2 * 2^-14
#define GSTRIDE 512         // floats per block in gram scratch (ReG 256 + ImG 256)

// ---------------------------------------------------------------------------
// Kernel 1 (parallel, all WGPs): per-block complex Gram tiles via f32 WMMA.
// ---------------------------------------------------------------------------
__global__ __launch_bounds__(32, 1)
void cma_gram(const float* __restrict__ fre,
              const float* __restrict__ fim,
              float* __restrict__ gram)
{
  __shared__ float Fls[BLK][LSTR];   // interleaved (re,im) features
  __shared__ float Gls[BLK][LSTR];   // rotated (-im,re) features

  const int lane = threadIdx.x;
  const int r16  = lane & 15;
  const int half = lane >> 4;
  const int b    = blockIdx.x;
  const size_t base = (size_t)b * BLK * QF;

  #pragma unroll
  for (int it = 0; it < 15; ++it) {
    const int idx = it * 32 + lane;        // 0..479 == m*QF + q
    const int m = idx / QF;
    const int q = idx - m * QF;
    const float re = fre[base + idx];
    const float im = fim[base + idx];
    Fls[m][2*q]   = re;   Fls[m][2*q+1] = im;
    Gls[m][2*q]   = -im;  Gls[m][2*q+1] = re;
  }
  __syncthreads();

  v8f accRe = {0,0,0,0,0,0,0,0};
  v8f accIm = {0,0,0,0,0,0,0,0};
  const int cofs = 2 * half;
  for (int kc = 0; kc < 30; ++kc) {
    const int col = 4 * kc + cofs;
    const v2f bF = *(const v2f*)&Fls[r16][col];
    const v2f aG = *(const v2f*)&Gls[r16][col];
    accRe = __builtin_amdgcn_wmma_f32_16x16x4_f32(false, bF, false, bF,
                                                  (short)0, accRe, false, false);
    accIm = __builtin_amdgcn_wmma_f32_16x16x4_f32(false, aG, false, bF,
                                                  (short)0, accIm, false, false);
  }

  float* g = gram + (size_t)b * GSTRIDE;   // store in D-register layout
  #pragma unroll
  for (int r = 0; r < 8; ++r) {
    g[r * 32 + lane]       = accRe[r];
    g[256 + r * 32 + lane] = accIm[r];
  }
}

// ---------------------------------------------------------------------------
// Kernel 2 (single wave): w-dependent filter GEMM + recurrence + tap update.
// ---------------------------------------------------------------------------
__global__ __launch_bounds__(32, 1)
void cma_scan_lite(const float* __restrict__ fre,
                   const float* __restrict__ fim,
                   const float* __restrict__ gram,
                   float* __restrict__ out)
{
  __shared__ float Fls[BLK][LSTR];   // interleaved (re,im) features
  __shared__ float Wls[BLK][LSTR];   // rows 0..3 = real-structured weight rows
  __shared__ float Cls[2][BLK][2];   // per-step complex update coefficients

  const int lane = threadIdx.x;
  const int r16  = lane & 15;
  const int half = lane >> 4;

  for (int idx = lane; idx < BLK * LSTR; idx += 32)
    (&Wls[0][0])[idx] = 0.0f;
  __syncthreads();
  if (lane < 2) {                          // central spike w[i][i][7] = 1
    const int i = lane, q = 14 + i;
    Wls[2*i][2*q]     = 1.0f;
    Wls[2*i+1][2*q+1] = 1.0f;
  }
  __syncthreads();

  for (int b = 0; b < NBLK; ++b) {
    const size_t base = (size_t)b * BLK * QF;
    const float* g = gram + (size_t)b * GSTRIDE;

    // Gram tile loads first (independent of LDS; L2-resident, overlap WMMA)
    float gr[8], gi[8];
    #pragma unroll
    for (int r = 0; r < 8; ++r) {
      gr[r] = g[r * 32 + lane];
      gi[r] = g[256 + r * 32 + lane];
    }

    // stage features
    #pragma unroll
    for (int it = 0; it < 15; ++it) {
      const int idx = it * 32 + lane;
      const int m = idx / QF;
      const int q = idx - m * QF;
      Fls[m][2*q]   = fre[base + idx];
      Fls[m][2*q+1] = fim[base + idx];
    }
    if (b + 1 < NBLK) {                    // WGP-scope prefetch of next block
      __builtin_prefetch(fre + base + BLK*QF + lane*16, 0, 3);
      __builtin_prefetch(fim + base + BLK*QF + lane*16, 0, 3);
      __builtin_prefetch(g + GSTRIDE + lane*16, 0, 3);
    }
    __syncthreads();

    // P = Wr * F^T  (M16 x N16 x K120 -> 30 f32 WMMAs; rows 0..3 meaningful)
    v8f accP = {0,0,0,0,0,0,0,0};
    const int cofs = 2 * half;
    for (int kc = 0; kc < 30; ++kc) {
      const int col = 4 * kc + cofs;
      const v2f bF = *(const v2f*)&Fls[r16][col];
      const v2f aW = *(const v2f*)&Wls[r16][col];
      accP = __builtin_amdgcn_wmma_f32_16x16x4_f32(false, aW, false, bF,
                                                   (short)0, accP, false, false);
    }

    float vre, vim;
    {
      const float a0 = __shfl(accP[0], r16, 32);   // Re v0[n]
      const float a1 = __shfl(accP[1], r16, 32);   // Im v0[n]
      const float a2 = __shfl(accP[2], r16, 32);   // Re v1[n]
      const float a3 = __shfl(accP[3], r16, 32);   // Im v1[n]
      vre = half ? a2 : a0;
      vim = half ? a3 : a1;
    }

    // 16-step serial recurrence
    float ore = 0.0f, oim = 0.0f;
    #pragma unroll
    for (int m = 0; m < BLK; ++m) {
      const float vmre = __shfl(vre, (lane & 16) + m, 32);
      const float vmim = __shfl(vim, (lane & 16) + m, 32);
      const float e = R2CMA - (vmre*vmre + vmim*vmim);
      const float s = (e > 0.0f) ? 1.0f : ((e < 0.0f) ? -1.0f : 0.0f);
      const float cre = MU * s * vmre;
      const float cim = MU * s * vmim;
      if (r16 == m) {
        ore = vmre; oim = vmim;
        Cls[half][m][0] = cre; Cls[half][m][1] = cim;
      }
      const int srcg = (m < 8) ? r16 : (16 + r16);
      const float gre = __shfl(gr[m & 7], srcg, 32);
      const float gim = __shfl(gi[m & 7], srcg, 32);
      vre += cre * gre - cim * gim;
      vim += cre * gim + cim * gre;
    }

    {
      const size_t o = (((size_t)(b * BLK + r16)) * 2 + half) * 2;
      out[o]     = ore;
      out[o + 1] = oim;
    }
    __syncthreads();

    // exact end-of-block weight update
    if (lane < QF) {
      const int q2 = 2 * lane;
      #pragma unroll
      for (int i = 0; i < 2; ++i) {
        float dre = 0.0f, dimv = 0.0f;
        #pragma unroll
        for (int m = 0; m < BLK; ++m) {
          const float cr = Cls[i][m][0];
          const float ci = Cls[i][m][1];
          const float ur = Fls[m][q2];
          const float ui = Fls[m][q2 + 1];
          dre  += cr * ur + ci * ui;
          dimv += ci * ur - cr * ui;
        }
        Wls[2*i][q2]       += dre;
        Wls[2*i][q2 + 1]   -= dimv;
        Wls[2*i+1][q2]     += dimv;
        Wls[2*i+1][q2 + 1] += dre;
      }
    }
    __syncthreads();
  }
}

// ---------------------------------------------------------------------------
// Fallback: monolithic single-wave kernel (Gram + P + scan all inline).
// ---------------------------------------------------------------------------
__global__ __launch_bounds__(32, 1)
void cma_block_exact(const float* __restrict__ fre,
                     const float* __restrict__ fim,
                     float* __restrict__ out)
{
  __shared__ float Fls[BLK][LSTR];
  __shared__ float Gls[BLK][LSTR];
  __shared__ float Wls[BLK][LSTR];
  __shared__ float Cls[2][BLK][2];

  const int lane = threadIdx.x;
  const int r16  = lane & 15;
  const int half = lane >> 4;

  for (int idx = lane; idx < BLK * LSTR; idx += 32)
    (&Wls[0][0])[idx] = 0.0f;
  __syncthreads();
  if (lane < 2) {
    const int i = lane, q = 14 + i;
    Wls[2*i][2*q]     = 1.0f;
    Wls[2*i+1][2*q+1] = 1.0f;
  }
  __syncthreads();

  for (int b = 0; b < NBLK; ++b) {
    const size_t base = (size_t)b * BLK * QF;

    #pragma unroll
    for (int it = 0; it < 15; ++it) {
      const int idx = it * 32 + lane;
      const int m = idx / QF;
      const int q = idx - m * QF;
      const float re = fre[base + idx];
      const float im = fim[base + idx];
      Fls[m][2*q]   = re;   Fls[m][2*q+1] = im;
      Gls[m][2*q]   = -im;  Gls[m][2*q+1] = re;
    }
    if (b + 1 < NBLK) {
      __builtin_prefetch(fre + base + BLK*QF + lane*16, 0, 3);
      __builtin_prefetch(fim + base + BLK*QF + lane*16, 0, 3);
    }
    __syncthreads();

    v8f accRe = {0,0,0,0,0,0,0,0};
    v8f accIm = {0,0,0,0,0,0,0,0};
    v8f accP  = {0,0,0,0,0,0,0,0};
    const int cofs = 2 * half;
    for (int kc = 0; kc < 30; ++kc) {
      const int col = 4 * kc + cofs;
      const v2f bF = *(const v2f*)&Fls[r16][col];
      const v2f aG = *(const v2f*)&Gls[r16][col];
      const v2f aW = *(const v2f*)&Wls[r16][col];
      accRe = __builtin_amdgcn_wmma_f32_16x16x4_f32(false, bF, false, bF,
                                                    (short)0, accRe, false, false);
      accIm = __builtin_amdgcn_wmma_f32_16x16x4_f32(false, aG, false, bF,
                                                    (short)0, accIm, false, false);
      accP  = __builtin_amdgcn_wmma_f32_16x16x4_f32(false, aW, false, bF,
                                                    (short)0, accP,  false, false);
    }

    float vre, vim;
    {
      const float a0 = __shfl(accP[0], r16, 32);
      const float a1 = __shfl(accP[1], r16, 32);
      const float a2 = __shfl(accP[2], r16, 32);
      const float a3 = __shfl(accP[3], r16, 32);
      vre = half ? a2 : a0;
      vim = half ? a3 : a1;
    }

    float ore = 0.0f, oim = 0.0f;
    #pragma unroll
    for (int m = 0; m < BLK; ++m) {
      const float vmre = __shfl(vre, (lane & 16) + m, 32);
      const float vmim = __shfl(vim, (lane & 16) + m, 32);
      const float e = R2CMA - (vmre*vmre + vmim*vmim);
      const float s = (e > 0.0f) ? 1.0f : ((e < 0.0f) ? -1.0f : 0.0f);
      const float cre = MU * s * vmre;
      const float cim = MU * s * vmim;
      if (r16 == m) {
        ore = vmre; oim = vmim;
        Cls[half][m][0] = cre; Cls[half][m][1] = cim;
      }
      const int srcg = (m < 8) ? r16 : (16 + r16);
      const float gre = __shfl(accRe[m & 7], srcg, 32);
      const float gim = __shfl(accIm[m & 7], srcg, 32);
      vre += cre * gre - cim * gim;
      vim += cre * gim + cim * gre;
    }

    {
      const size_t o = (((size_t)(b * BLK + r16)) * 2 + half) * 2;
      out[o]     = ore;
      out[o + 1] = oim;
    }
    __syncthreads();

    if (lane < QF) {
      const int q2 = 2 * lane;
      #pragma unroll
      for (int i = 0; i < 2; ++i) {
        float dre = 0.0f, dimv = 0.0f;
        #pragma unroll
        for (int m = 0; m < BLK; ++m) {
          const float cr = Cls[i][m][0];
          const float ci = Cls[i][m][1];
          const float ur = Fls[m][q2];
          const float ui = Fls[m][q2 + 1];
          dre  += cr * ur + ci * ui;
          dimv += ci * ur - cr * ui;
        }
        Wls[2*i][q2]       += dre;
        Wls[2*i][q2 + 1]   -= dimv;
        Wls[2*i+1][q2]     += dimv;
        Wls[2*i+1][q2 + 1] += dre;
      }
    }
    __syncthreads();
  }
}

extern "C" void kernel_launch(void* const* d_in, const int* in_sizes, int n_in,
                              void* d_out, int out_size, void* d_ws, size_t ws_size,
                              hipStream_t stream) {
  (void)in_sizes; (void)n_in; (void)out_size;
  const float* fre = (const float*)d_in[0];
  const float* fim = (const float*)d_in[1];
  float* out = (float*)d_out;

  const size_t gram_bytes = (size_t)NBLK * GSTRIDE * sizeof(float);  // 16 MB
  if (ws_size >= gram_bytes) {
    // Parallel pre-pass over all WGPs (w-independent Gram tiles), then the
    // shortened serial chain.
    hipLaunchKernelGGL(cma_gram, dim3(NBLK), dim3(32), 0, stream,
                       fre, fim, (float*)d_ws);
    hipLaunchKernelGGL(cma_scan_lite, dim3(1), dim3(32), 0, stream,
                       fre, fim, (const float*)d_ws, out);
  } else {
    hipLaunchKernelGGL(cma_block_exact, dim3(1), dim3(32), 0, stream,
                       fre, fim, out);
  }
}